// AllAtomAtomAttention_78142634983967
// MI455X (gfx1250) — compile-verified
//
#include <hip/hip_runtime.h>
#include <math.h>

typedef __attribute__((ext_vector_type(2))) float v2f;
typedef __attribute__((ext_vector_type(8))) float v8f;

#define LATENT_D   128
#define N_HEADS_D  4
#define HEAD_D     32
#define NE_D       32
#define EDIM_D     16
#define RBF_D      16
#define ZEMB_D     32
#define SCORE_SCL  0.17677669529663687f   /* 32^-0.5 */
#define KQ_D       144                    /* 128 + 16 */
#define KP_D       180                    /* 128 + 32 + 1 + 16 padded to mult-of-4 */
#define KP_REAL    177

// ---------------------------------------------------------------------------
// Generic WMMA GEMM:  C[M x 128] = act(A[M x K] @ W[K x 128] + bias) * rowScale
// One wave32 computes one 16x16 tile via V_WMMA_F32_16X16X4_F32.
// Block = 256 threads = 8 waves covering all 128 output columns; grid.x = M/16.
// A tile (16 x K) is staged in LDS and shared by the 8 waves.
// ---------------------------------------------------------------------------
template<int SILU>
__global__ __launch_bounds__(256)
void gemm128_wmma(const float* __restrict__ A, const float* __restrict__ W,
                  const float* __restrict__ bias, const float* __restrict__ rowScale,
                  float* __restrict__ C, int K)
{
    __shared__ float As[16 * 192];
    const int m0  = blockIdx.x * 16;
    const int tid = threadIdx.x;

    for (int i = tid; i < 16 * K; i += 256)
        As[i] = A[(size_t)(m0 + i / K) * K + (i % K)];
    __syncthreads();

    const int wave = tid >> 5;
    const int lane = tid & 31;
    const int mrow = lane & 15;     // M index within tile (lanes 0-15 / 16-31)
    const int hi   = lane >> 4;     // half-wave selector
    const int n    = wave * 16 + (lane & 15);

    v8f acc = {0.f, 0.f, 0.f, 0.f, 0.f, 0.f, 0.f, 0.f};
    for (int k = 0; k < K; k += 4) {
        v2f a, b;
        // A 16x4 layout: VGPR0 = K 0|2, VGPR1 = K 1|3 (lanes 0-15 | 16-31)
        a.x = As[mrow * K + k + 2 * hi];
        a.y = As[mrow * K + k + 2 * hi + 1];
        // B 4x16 layout: lane holds column n; rows K 0|2 and 1|3
        b.x = W[(size_t)(k + 2 * hi)     * 128 + n];
        b.y = W[(size_t)(k + 2 * hi + 1) * 128 + n];
        acc = __builtin_amdgcn_wmma_f32_16x16x4_f32(
                  false, a, false, b, (short)0, acc, false, false);
    }

    const float bn = bias[n];
#pragma unroll
    for (int r = 0; r < 8; ++r) {
        int   row = m0 + r + 8 * hi;      // D layout: VGPR r -> M = r (+8 for hi half)
        float v   = acc[r] + bn;
        if (SILU) v = v / (1.0f + expf(-v));
        if (rowScale) v *= rowScale[row];
        C[(size_t)row * 128 + n] = v;
    }
}

// ---------------------------------------------------------------------------
// Build query-MLP input: rows = flat*nE, cols = 144 = [h_atom | e_feat]
// ---------------------------------------------------------------------------
__global__ void build_q_input(const float* __restrict__ h, const float* __restrict__ efeat,
                              float* __restrict__ Xq, int total)
{
    int idx = blockIdx.x * blockDim.x + threadIdx.x;
    if (idx >= total) return;
    int row = idx / KQ_D, col = idx % KQ_D;
    int a = row >> 5, e = row & 31;
    Xq[idx] = (col < 128) ? h[(size_t)a * 128 + col]
                          : efeat[e * EDIM_D + (col - 128)];
}

// ---------------------------------------------------------------------------
// Build pair_static (E x 180): [h_dst | z_emb[z_dst] | is_self | rbf | pad0];
// also writes cosine-cutoff(dist) once per edge.
// ---------------------------------------------------------------------------
__global__ void build_pair(const float* __restrict__ h, const int* __restrict__ z,
                           const int* __restrict__ asrc, const int* __restrict__ adst,
                           const float* __restrict__ dist, const float* __restrict__ zemb,
                           float* __restrict__ Xp, float* __restrict__ cut, int total)
{
    int idx = blockIdx.x * blockDim.x + threadIdx.x;
    if (idx >= total) return;
    int row = idx / KP_D, col = idx % KP_D;
    int dst = adst[row];
    float v;
    if (col < 128) {
        v = h[(size_t)dst * 128 + col];
    } else if (col < 160) {
        v = zemb[(size_t)z[dst] * ZEMB_D + (col - 128)];
    } else if (col == 160) {
        v = (asrc[row] == dst) ? 1.0f : 0.0f;
    } else if (col < 177) {
        int   j  = col - 161;
        float cc = 5.0f * (float)j / 15.0f;          // linspace(0,5,16)
        float dd = dist[row] - cc;
        v = expf(-4.5f * dd * dd);                    // coeff = -0.5/width^2, width=1/3
    } else {
        v = 0.0f;
    }
    Xp[idx] = v;
    if (col == 0) {
        float d = dist[row];
        cut[row] = 0.5f * (cosf(3.14159265358979323846f * d / 5.0f) + 1.0f)
                 * (d < 5.0f ? 1.0f : 0.0f);
    }
}

// Pad a (KrealxN) weight to (KpadxN) with zero rows
__global__ void pad_weight(const float* __restrict__ W, float* __restrict__ Wp,
                           int Kreal, int total)
{
    int idx = blockIdx.x * blockDim.x + threadIdx.x;
    if (idx >= total) return;
    int r = idx / 128, c = idx % 128;
    Wp[idx] = (r < Kreal) ? W[(size_t)r * 128 + c] : 0.0f;
}

// ---------------------------------------------------------------------------
// Per-atom scatter softmax + weighted sum. One block = one query atom.
// Deterministic edge list via per-thread counts + LDS scan (no atomics).
// ---------------------------------------------------------------------------
#define MAX_EPA 2048
__global__ __launch_bounds__(256)
void attention_kernel(const float* __restrict__ Q, const float* __restrict__ Kb,
                      const float* __restrict__ Vb, const int* __restrict__ asrc,
                      const int* __restrict__ adst, const unsigned char* __restrict__ mask,
                      float* __restrict__ out, int E)
{
    const int a   = blockIdx.x;
    const int tid = threadIdx.x;

    __shared__ float q[NE_D * 128];
    __shared__ float accs[NE_D * 128];
    __shared__ float kv[256];          // [0..127] = k_e, [128..255] = v_e
    __shared__ float attn[128];
    __shared__ int   elist[MAX_EPA];
    __shared__ int   cnts[257];

    // ---- gather this atom's edges, order-preserving ----
    const int chunk = (E + 255) / 256;
    const int base  = tid * chunk;
    int cnt = 0;
    for (int i = 0; i < chunk; ++i) {
        int e = base + i;
        if (e < E && asrc[e] == a) cnt++;
    }
    cnts[tid + 1] = cnt;
    if (tid == 0) cnts[0] = 0;
    __syncthreads();
    for (int s = 1; s < 256; s <<= 1) {
        int v = cnts[tid + 1] + ((tid >= s) ? cnts[tid + 1 - s] : 0);
        __syncthreads();
        cnts[tid + 1] = v;
        __syncthreads();
    }
    int w = cnts[tid];
    for (int i = 0; i < chunk; ++i) {
        int e = base + i;
        if (e < E && asrc[e] == a) { if (w < MAX_EPA) elist[w] = e; w++; }
    }
    for (int i = tid; i < NE_D * 128; i += 256) {
        q[i]    = Q[(size_t)a * (NE_D * 128) + i];
        accs[i] = 0.0f;
    }
    __syncthreads();
    const int nedge = (cnts[256] < MAX_EPA) ? cnts[256] : MAX_EPA;
    const unsigned char ma = mask[a];

    // ---- pass 1: online max / denom per (nE, head), lanes 0..127 ----
    float mval = -1e30f, lval = 0.0f;
    const int neI = tid >> 2, hdI = tid & 3;
    for (int ei = 0; ei < nedge; ++ei) {
        int e = elist[ei];
        if (ma && mask[adst[e]]) {
            if (tid < 128) kv[tid] = Kb[(size_t)e * 128 + tid];
            __syncthreads();
            if (tid < 128) {
                const float* qp = &q[neI * 128 + hdI * HEAD_D];
                const float* kp = &kv[hdI * HEAD_D];
                float s = 0.0f;
                for (int d = 0; d < HEAD_D; ++d) s += qp[d] * kp[d];
                s *= SCORE_SCL;
                float mn = fmaxf(mval, s);
                lval = lval * expf(mval - mn) + expf(s - mn);
                mval = mn;
            }
            __syncthreads();
        }
    }
    const float inv = (tid < 128) ? 1.0f / fmaxf(lval, 1e-30f) : 0.0f;

    // ---- pass 2: attn weights + acc += attn * v ----
    for (int ei = 0; ei < nedge; ++ei) {
        int e = elist[ei];
        if (ma && mask[adst[e]]) {
            if (tid < 128) kv[tid]       = Kb[(size_t)e * 128 + tid];
            else           kv[tid]       = Vb[(size_t)e * 128 + (tid - 128)];
            __syncthreads();
            if (tid < 128) {
                const float* qp = &q[neI * 128 + hdI * HEAD_D];
                const float* kp = &kv[hdI * HEAD_D];
                float s = 0.0f;
                for (int d = 0; d < HEAD_D; ++d) s += qp[d] * kp[d];
                s *= SCORE_SCL;
                attn[tid] = expf(s - mval) * inv;
            }
            __syncthreads();
            const int b0 = tid * 16;
#pragma unroll
            for (int j = 0; j < 16; ++j) {
                int idx = b0 + j;
                int ne  = idx >> 7;
                int d   = idx & 127;
                accs[idx] += attn[(ne << 2) + (d >> 5)] * kv[128 + d];
            }
            __syncthreads();
        }
    }
    for (int i = tid; i < NE_D * 128; i += 256)
        out[(size_t)a * (NE_D * 128) + i] = accs[i];
}

// ---------------------------------------------------------------------------
extern "C" void kernel_launch(void* const* d_in, const int* in_sizes, int n_in,
                              void* d_out, int out_size, void* d_ws, size_t ws_size,
                              hipStream_t stream)
{
    const float*         h     = (const float*)d_in[0];
    const int*           z     = (const int*)d_in[1];
    const unsigned char* mask  = (const unsigned char*)d_in[2];
    const float*         efeat = (const float*)d_in[3];
    const int*           asrc  = (const int*)d_in[4];
    const int*           adst  = (const int*)d_in[5];
    const float*         dist  = (const float*)d_in[6];

    const int flat = in_sizes[0] / 128;    // 512
    const int E    = in_sizes[4];          // 8192
    const int Mq   = flat * NE_D;          // 16384

    // Resolve param ordering: pytree dicts are key-sorted (key_mlp, out_proj,
    // query_mlp, value_mlp, z_emb); fall back to insertion order if slot 7
    // is z_emb (101*32 = 3232 elements).
    const float *zemb, *Wq[3], *bq[3], *Wk[3], *bk[3], *Wv[3], *bv[3], *Wo[2], *bo[2];
    if (in_sizes[7] == 3232) {   // insertion order: z_emb, query, key, value, out
        zemb = (const float*)d_in[7];
        int p = 8;
        for (int i = 0; i < 3; ++i) { Wq[i] = (const float*)d_in[p++]; bq[i] = (const float*)d_in[p++]; }
        for (int i = 0; i < 3; ++i) { Wk[i] = (const float*)d_in[p++]; bk[i] = (const float*)d_in[p++]; }
        for (int i = 0; i < 3; ++i) { Wv[i] = (const float*)d_in[p++]; bv[i] = (const float*)d_in[p++]; }
        for (int i = 0; i < 2; ++i) { Wo[i] = (const float*)d_in[p++]; bo[i] = (const float*)d_in[p++]; }
    } else {                      // sorted order: key, out_proj, query, value, z_emb
        int p = 7;
        for (int i = 0; i < 3; ++i) { Wk[i] = (const float*)d_in[p++]; bk[i] = (const float*)d_in[p++]; }
        for (int i = 0; i < 2; ++i) { Wo[i] = (const float*)d_in[p++]; bo[i] = (const float*)d_in[p++]; }
        for (int i = 0; i < 3; ++i) { Wq[i] = (const float*)d_in[p++]; bq[i] = (const float*)d_in[p++]; }
        for (int i = 0; i < 3; ++i) { Wv[i] = (const float*)d_in[p++]; bv[i] = (const float*)d_in[p++]; }
        zemb = (const float*)d_in[p];
    }

    // Workspace layout (floats). region0 is reused sequentially:
    // Xq -> Xp -> out_flat (lifetimes do not overlap).
    float* W0  = (float*)d_ws;
    size_t o0 = 0;                                      // 2,359,296 (16384*144)
    size_t o1 = o0 + (size_t)Mq * KQ_D;                 // H1: 16384*128
    size_t o2 = o1 + (size_t)Mq * 128;                  // H2: 16384*128
    size_t o3 = o2 + (size_t)Mq * 128;                  // Qatom: 16384*128
    size_t o4 = o3 + (size_t)Mq * 128;                  // Kbuf: 8192*128
    size_t o5 = o4 + (size_t)E * 128;                   // Vbuf: 8192*128
    size_t o6 = o5 + (size_t)E * 128;                   // cutoff: E
    size_t o7 = o6 + (size_t)E;                         // Wk0 padded: 180*128
    size_t o8 = o7 + (size_t)KP_D * 128;                // Wv0 padded: 180*128
    float* Xq      = W0 + o0;   // also Xp, also out_flat
    float* H1      = W0 + o1;
    float* H2      = W0 + o2;
    float* Qatom   = W0 + o3;
    float* Kbuf    = W0 + o4;
    float* Vbuf    = W0 + o5;
    float* cutoff  = W0 + o6;
    float* Wk0p    = W0 + o7;
    float* Wv0p    = W0 + o8;
    (void)ws_size; (void)n_in; (void)out_size;

    // ---- query path (deduped: unique atoms x e_feat rows) ----
    {
        int total = Mq * KQ_D;
        build_q_input<<<(total + 255) / 256, 256, 0, stream>>>(h, efeat, Xq, total);
        gemm128_wmma<1><<<Mq / 16, 256, 0, stream>>>(Xq, Wq[0], bq[0], nullptr, H1, KQ_D);
        gemm128_wmma<1><<<Mq / 16, 256, 0, stream>>>(H1, Wq[1], bq[1], nullptr, H2, 128);
        gemm128_wmma<0><<<Mq / 16, 256, 0, stream>>>(H2, Wq[2], bq[2], nullptr, Qatom, 128);
    }

    // ---- key / value path ----
    {
        int total = E * KP_D;
        build_pair<<<(total + 255) / 256, 256, 0, stream>>>(h, z, asrc, adst, dist, zemb,
                                                            Xq /*as Xp*/, cutoff, total);
        int wtot = KP_D * 128;
        pad_weight<<<(wtot + 255) / 256, 256, 0, stream>>>(Wk[0], Wk0p, KP_REAL, wtot);
        pad_weight<<<(wtot + 255) / 256, 256, 0, stream>>>(Wv[0], Wv0p, KP_REAL, wtot);

        gemm128_wmma<1><<<E / 16, 256, 0, stream>>>(Xq, Wk0p, bk[0], nullptr, H1, KP_D);
        gemm128_wmma<1><<<E / 16, 256, 0, stream>>>(H1, Wk[1], bk[1], nullptr, H2, 128);
        gemm128_wmma<0><<<E / 16, 256, 0, stream>>>(H2, Wk[2], bk[2], nullptr, Kbuf, 128);

        gemm128_wmma<1><<<E / 16, 256, 0, stream>>>(Xq, Wv0p, bv[0], nullptr, H1, KP_D);
        gemm128_wmma<1><<<E / 16, 256, 0, stream>>>(H1, Wv[1], bv[1], nullptr, H2, 128);
        gemm128_wmma<0><<<E / 16, 256, 0, stream>>>(H2, Wv[2], bv[2], cutoff, Vbuf, 128);
    }

    // ---- scatter softmax attention (region0 reused as out_flat) ----
    attention_kernel<<<flat, 256, 0, stream>>>(Qatom, Kbuf, Vbuf, asrc, adst, mask,
                                               Xq /*as out_flat*/, E);

    // ---- out_proj ----
    gemm128_wmma<1><<<Mq / 16, 256, 0, stream>>>(Xq, Wo[0], bo[0], nullptr, H2, 128);
    gemm128_wmma<0><<<Mq / 16, 256, 0, stream>>>(H2, Wo[1], bo[1], nullptr, (float*)d_out, 128);
}